// RegressionModel_65481071402839
// MI455X (gfx1250) — compile-verified
//
#include <hip/hip_runtime.h>

// ---------------------------------------------------------------------------
// 16-qubit circuit regression for MI455X (gfx1250, wave32, WMMA).
//
// Math: per-wire gates fold into 4 Kronecker 16x16 complex unitaries G0..G3
// (one per index nibble; wire i <-> bit 15-i). CNOT chain is a permutation
// folded into readout weights d[j] = sum_n w_n*(1-2*parity(j>>(15-n))).
// out[b] = (sum_j |phi_j|^2 d_j) / ||psi||^2 + head_b, phi = (G0xG1xG2xG3) psi.
//
// pass1: apply G3 (bits 3:0) then G2 (bits 7:4) per contiguous 256-amp tile,
//        accumulate norm^2, write intermediate [b][h*256+l].
// pass2: apply G1 (bits 11:8) then G0 (bits 15:12) per l-column tile, weight
//        by d-table, reduce to accum[b].
// Each 16x16x16 complex matmul = 4 real matmuls = 16 x V_WMMA_F32_16X16X4_F32.
// ---------------------------------------------------------------------------

#define QDIM 65536
#define QBATCH 512

typedef __attribute__((ext_vector_type(2))) float v2f;
typedef __attribute__((ext_vector_type(8))) float v8f;

__device__ __forceinline__ v8f wmma4(v2f a, v2f b, v8f c) {
  // D = A(16x4) * B(4x16) + C ; mirrors 8-arg pattern of the f16/bf16 WMMAs.
  return __builtin_amdgcn_wmma_f32_16x16x4_f32(
      /*neg_a=*/false, a, /*neg_b=*/false, b,
      /*c_mod=*/(short)0, c, /*reuse_a=*/false, /*reuse_b=*/false);
}

// Chain 4 K-chunks -> full 16x16x16 real matmul accumulate.
__device__ __forceinline__ v8f mm16(const v2f a[4], const v2f b[4], v8f c) {
#pragma unroll
  for (int t = 0; t < 4; ++t) c = wmma4(a[t], b[t], c);
  return c;
}

// Fragment element mapping (f32 WMMA, wave32):
//   A 16x4 chunk t: lane -> M=lane%16, K = 4t + j + 2*(lane/16)  (j = vgpr 0/1)
//   B 4x16 chunk t: lane -> N=lane%16, K = 4t + j + 2*(lane/16)  (mirrored)
//   C/D: vgpr v -> M = v + 8*(lane/16), N = lane%16
// load_rowfrag: element = src[lm*stride + (4t+j+2lh)]   (row-major M x K)
__device__ __forceinline__ void load_rowfrag(const float* src, int stride,
                                             int lm, int lh, v2f f[4]) {
  const float* p = src + lm * stride + 2 * lh;
#pragma unroll
  for (int t = 0; t < 4; ++t) { f[t][0] = p[4 * t]; f[t][1] = p[4 * t + 1]; }
}
// load_colfrag: element = src[(4t+j+2lh)*stride + lm]   (row-major K x N)
__device__ __forceinline__ void load_colfrag(const float* src, int stride,
                                             int lm, int lh, v2f f[4]) {
  const float* p = src + 2 * lh * stride + lm;
#pragma unroll
  for (int t = 0; t < 4; ++t) {
    f[t][0] = p[(4 * t) * stride];
    f[t][1] = p[(4 * t + 1) * stride];
  }
}
// store D fragment as row-major 16x16 with padded stride
__device__ __forceinline__ void store_dfrag(float* dst, int stride,
                                            int lm, int lh, v8f d) {
  float* p = dst + (8 * lh) * stride + lm;
#pragma unroll
  for (int v = 0; v < 8; ++v) p[v * stride] = d[v];
}

// ---------------------------------------------------------------------------
// prep: build per-wire 2x2 unitaries, Kronecker them into 4 16x16 complex
// gate matrices, build the 64K readout weight table, zero accumulators.
// ---------------------------------------------------------------------------
__global__ __launch_bounds__(256) void prep_kernel(
    const float* __restrict__ params, const float* __restrict__ head_w,
    float* __restrict__ gmat, float* __restrict__ dtab,
    float* __restrict__ norm2, float* __restrict__ accum) {
  __shared__ float ur[16][2][2], ui[16][2][2];
  const int tid = threadIdx.x;
  if (tid < 16) {
    const int i = tid;
    const float hy = 0.5f * params[3 * i + 0];
    const float hx = 0.5f * params[3 * i + 1];
    const float hz = 0.5f * params[3 * i + 2];
    const float cy = cosf(hy), sy = sinf(hy);
    const float cx = cosf(hx), sx = sinf(hx);
    const float cz = cosf(hz), sz = sinf(hz);
    // M = Rx*Ry
    const float m00r = cx * cy, m00i = -sx * sy;
    const float m01r = -cx * sy, m01i = -sx * cy;
    const float m10r = cx * sy, m10i = -sx * cy;
    const float m11r = cx * cy, m11i = sx * sy;
    // U = Rz*M : row0 *= (cz - i sz), row1 *= (cz + i sz)
    ur[i][0][0] = cz * m00r + sz * m00i; ui[i][0][0] = cz * m00i - sz * m00r;
    ur[i][0][1] = cz * m01r + sz * m01i; ui[i][0][1] = cz * m01i - sz * m01r;
    ur[i][1][0] = cz * m10r - sz * m10i; ui[i][1][0] = cz * m10i + sz * m10r;
    ur[i][1][1] = cz * m11r - sz * m11i; ui[i][1][1] = cz * m11i + sz * m11r;
  }
  __syncthreads();
  // Kronecker: group g covers wires 4g..4g+3, wire 4g = MSB of the nibble.
  const int r = tid >> 4, c = tid & 15;
#pragma unroll
  for (int g = 0; g < 4; ++g) {
    float pr = 1.f, pi = 0.f;
#pragma unroll
    for (int q = 0; q < 4; ++q) {
      const int w = 4 * g + q;
      const int rb = (r >> (3 - q)) & 1, cb = (c >> (3 - q)) & 1;
      const float qr = ur[w][rb][cb], qi = ui[w][rb][cb];
      const float nr = pr * qr - pi * qi;
      const float ni = pr * qi + pi * qr;
      pr = nr; pi = ni;
    }
    gmat[g * 512 + r * 16 + c] = pr;         // real
    gmat[g * 512 + 256 + r * 16 + c] = pi;   // imag
  }
  // Readout weights with CNOT-chain permutation folded in (prefix parity).
  float w[16];
#pragma unroll
  for (int n = 0; n < 16; ++n) w[n] = head_w[n];
  for (int k = 0; k < 256; ++k) {
    const int j = tid * 256 + k;
    float d = 0.f;
#pragma unroll
    for (int n = 0; n < 16; ++n)
      d += w[n] * (1.0f - 2.0f * (float)(__popc(j >> (15 - n)) & 1));
    dtab[j] = d;
  }
  norm2[tid] = 0.f; norm2[tid + 256] = 0.f;
  accum[tid] = 0.f; accum[tid + 256] = 0.f;
}

// ---------------------------------------------------------------------------
// pass1: low byte. Tile = 256 contiguous amplitudes (l0,l1) at fixed h.
// T = V * G3^T ; V' = G2 * T. 8 waves * 4 tiles per workgroup.
// ---------------------------------------------------------------------------
__global__ __launch_bounds__(256) void pass1_kernel(
    const float* __restrict__ sr, const float* __restrict__ si,
    const float* __restrict__ gmat,
    float* __restrict__ outR, float* __restrict__ outI,
    float* __restrict__ norm2) {
  __shared__ float vst[8][2][16 * 17];   // per-wave V stage (r/i), stride 17
  __shared__ float tst[8][2][16 * 19];   // per-wave T stage (r/i), stride 19
  __shared__ float red[256];
  const int tid = threadIdx.x;
  const int wave = tid >> 5, lane = tid & 31;
  const int lm = lane & 15, lh = lane >> 4;
  const int b = blockIdx.y, hblk = blockIdx.x;

  v2f g2r[4], g2i[4], g2in[4], g3r[4], g3i[4], g3in[4];
  load_rowfrag(gmat + 2 * 512, 16, lm, lh, g2r);        // A-side: G2
  load_rowfrag(gmat + 2 * 512 + 256, 16, lm, lh, g2i);
  load_rowfrag(gmat + 3 * 512, 16, lm, lh, g3r);        // B-side: gives V*G3^T
  load_rowfrag(gmat + 3 * 512 + 256, 16, lm, lh, g3i);
#pragma unroll
  for (int t = 0; t < 4; ++t) { g2in[t] = -g2i[t]; g3in[t] = -g3i[t]; }

  float nrm = 0.f;
  float* vr = vst[wave][0]; float* vi = vst[wave][1];
  float* tr = tst[wave][0]; float* ti = tst[wave][1];

#pragma unroll 1
  for (int it = 0; it < 4; ++it) {
    const int h = hblk * 32 + wave * 4 + it;
    const size_t gbase = (size_t)b * QDIM + (size_t)h * 256;
#pragma unroll
    for (int k = 0; k < 8; ++k) {          // coalesced 32B/lane stage-in
      const int f = lane * 8 + k;
      const float xr = sr[gbase + f], xi = si[gbase + f];
      nrm += xr * xr + xi * xi;
      const int li = (f >> 4) * 17 + (f & 15);
      vr[li] = xr; vi[li] = xi;
    }
    __syncthreads();
    v2f ar[4], ai[4];
    load_rowfrag(vr, 17, lm, lh, ar);      // A = V (M=l0, K=m1)
    load_rowfrag(vi, 17, lm, lh, ai);
    v8f Tr = {}; Tr = mm16(ar, g3r, Tr); Tr = mm16(ai, g3in, Tr);
    v8f Ti = {}; Ti = mm16(ar, g3i, Ti); Ti = mm16(ai, g3r, Ti);
    store_dfrag(tr, 19, lm, lh, Tr);
    store_dfrag(ti, 19, lm, lh, Ti);
    __syncthreads();
    v2f btr[4], bti[4];
    load_colfrag(tr, 19, lm, lh, btr);     // B = T (K=m0, N=l1)
    load_colfrag(ti, 19, lm, lh, bti);
    v8f Or = {}; Or = mm16(g2r, btr, Or); Or = mm16(g2in, bti, Or);
    v8f Oi = {}; Oi = mm16(g2r, bti, Oi); Oi = mm16(g2i, btr, Oi);
#pragma unroll
    for (int v = 0; v < 8; ++v) {          // 64B-segment coalesced stores
      const int off = (v + 8 * lh) * 16 + lm;
      outR[gbase + off] = Or[v];
      outI[gbase + off] = Oi[v];
    }
    __syncthreads();
  }
  red[tid] = nrm;
  __syncthreads();
  for (int s = 128; s > 0; s >>= 1) {
    if (tid < s) red[tid] += red[tid + s];
    __syncthreads();
  }
  if (tid == 0) atomicAdd(&norm2[b], red[0]);
}

// ---------------------------------------------------------------------------
// pass2: high byte. Tile per low-index l: W[h0][h1]. T = W*G1^T ; W' = G0*T.
// Then weighted probability reduction with dtab. 16 l per workgroup.
// ---------------------------------------------------------------------------
__global__ __launch_bounds__(256) void pass2_kernel(
    const float* __restrict__ inR, const float* __restrict__ inI,
    const float* __restrict__ gmat, const float* __restrict__ dtab,
    float* __restrict__ accum) {
  __shared__ float stg[2][16 * 273];       // [r/i][l][h1*17 + h0]
  __shared__ float tst[8][2][16 * 19];
  __shared__ float red[256];
  const int tid = threadIdx.x;
  const int wave = tid >> 5, lane = tid & 31;
  const int lm = lane & 15, lh = lane >> 4;
  const int b = blockIdx.y, lblk = blockIdx.x;

  v2f g0r[4], g0i[4], g0in[4], g1r[4], g1i[4], g1in[4];
  load_rowfrag(gmat + 0 * 512, 16, lm, lh, g0r);        // A-side: G0
  load_rowfrag(gmat + 0 * 512 + 256, 16, lm, lh, g0i);
  load_rowfrag(gmat + 1 * 512, 16, lm, lh, g1r);        // B-side: W*G1^T
  load_rowfrag(gmat + 1 * 512 + 256, 16, lm, lh, g1i);
#pragma unroll
  for (int t = 0; t < 4; ++t) { g0in[t] = -g0i[t]; g1in[t] = -g1i[t]; }

  const size_t base = (size_t)b * QDIM + (size_t)lblk * 16;
#pragma unroll
  for (int k = 0; k < 16; ++k) {           // 64B-segment strided stage-in
    const int f = k * 256 + tid;
    const int h = f >> 4, l = f & 15;
    const size_t g = base + (size_t)h * 256 + l;
    const int li = l * 273 + (h & 15) * 17 + (h >> 4);
    stg[0][li] = inR[g];
    stg[1][li] = inI[g];
  }
  __syncthreads();

  float acc = 0.f;
  float* tr = tst[wave][0]; float* ti = tst[wave][1];
#pragma unroll 1
  for (int it = 0; it < 2; ++it) {
    const int l = wave * 2 + it;
    v2f ar[4], ai[4];
    load_colfrag(stg[0] + l * 273, 17, lm, lh, ar);  // A = W (M=h0, K=h1)
    load_colfrag(stg[1] + l * 273, 17, lm, lh, ai);
    v8f Tr = {}; Tr = mm16(ar, g1r, Tr); Tr = mm16(ai, g1in, Tr);
    v8f Ti = {}; Ti = mm16(ar, g1i, Ti); Ti = mm16(ai, g1r, Ti);
    store_dfrag(tr, 19, lm, lh, Tr);
    store_dfrag(ti, 19, lm, lh, Ti);
    __syncthreads();
    v2f btr[4], bti[4];
    load_colfrag(tr, 19, lm, lh, btr);
    load_colfrag(ti, 19, lm, lh, bti);
    v8f Or = {}; Or = mm16(g0r, btr, Or); Or = mm16(g0in, bti, Or);
    v8f Oi = {}; Oi = mm16(g0r, bti, Oi); Oi = mm16(g0i, btr, Oi);
    const int jl = lblk * 16 + l;
#pragma unroll
    for (int v = 0; v < 8; ++v) {
      const int h0 = v + 8 * lh;
      const int j = (h0 * 16 + lm) * 256 + jl;   // final basis index
      acc += (Or[v] * Or[v] + Oi[v] * Oi[v]) * dtab[j];
    }
    __syncthreads();
  }
  red[tid] = acc;
  __syncthreads();
  for (int s = 128; s > 0; s >>= 1) {
    if (tid < s) red[tid] += red[tid + s];
    __syncthreads();
  }
  if (tid == 0) atomicAdd(&accum[b], red[0]);
}

__global__ __launch_bounds__(256) void finalize_kernel(
    const float* __restrict__ accum, const float* __restrict__ norm2,
    const float* __restrict__ head_b, float* __restrict__ out) {
  const int b = blockIdx.x * 256 + threadIdx.x;
  if (b < QBATCH) out[b] = accum[b] / norm2[b] + head_b[0];
}

// ---------------------------------------------------------------------------
extern "C" void kernel_launch(void* const* d_in, const int* in_sizes, int n_in,
                              void* d_out, int out_size, void* d_ws,
                              size_t ws_size, hipStream_t stream) {
  (void)in_sizes; (void)n_in; (void)out_size; (void)ws_size;
  const float* sr     = (const float*)d_in[0];
  const float* si     = (const float*)d_in[1];
  const float* params = (const float*)d_in[2];
  const float* head_w = (const float*)d_in[3];
  const float* head_b = (const float*)d_in[4];
  float* out = (float*)d_out;

  float* ws = (float*)d_ws;
  const size_t N = (size_t)QBATCH * QDIM;   // 33,554,432
  float* interR = ws;
  float* interI = interR + N;
  float* gmat   = interI + N;               // 4 * 512 floats
  float* dtab   = gmat + 2048;              // 65536 floats
  float* norm2  = dtab + 65536;             // 512
  float* accum  = norm2 + 512;              // 512

  prep_kernel<<<1, 256, 0, stream>>>(params, head_w, gmat, dtab, norm2, accum);
  pass1_kernel<<<dim3(8, QBATCH), 256, 0, stream>>>(sr, si, gmat, interR,
                                                    interI, norm2);
  pass2_kernel<<<dim3(16, QBATCH), 256, 0, stream>>>(interR, interI, gmat,
                                                     dtab, accum);
  finalize_kernel<<<2, 256, 0, stream>>>(accum, norm2, head_b, out);
}